// GCN_34531537060237
// MI455X (gfx1250) — compile-verified
//
#include <hip/hip_runtime.h>

typedef __attribute__((ext_vector_type(2))) float v2f;
typedef __attribute__((ext_vector_type(8))) float v8f;

#define DIM 128

// ---------- degree / normalization ----------
__global__ void k_deg_init(float* deg, int n) {
    int i = blockIdx.x * blockDim.x + threadIdx.x;
    if (i < n) deg[i] = 1.0f;               // self-loop contributes 1
}

__global__ void k_deg_count(const int* __restrict__ col, float* deg, int e) {
    int i = blockIdx.x * blockDim.x + threadIdx.x;
    if (i < e) atomicAdd(&deg[col[i]], 1.0f);
}

__global__ void k_deg_rsqrt(float* deg, int n) {
    int i = blockIdx.x * blockDim.x + threadIdx.x;
    if (i < n) deg[i] = rsqrtf(deg[i]);     // deg >= 1 always (self loops)
}

// ---------- fp32 WMMA GEMM:  Y[n x 128] = X[n x 128] @ W[128 x 128] (+ cvec) ----------
// one wave computes one 16x16 tile; blockDim=256 (8 waves) covers all 128 columns
__global__ __launch_bounds__(256) void k_gemm128_wmma(
    const float* __restrict__ X, const float* __restrict__ W,
    const float* __restrict__ cvec, float* __restrict__ Y, int n) {
    const int lane = threadIdx.x & 31;
    const int wave = threadIdx.x >> 5;
    const int m0   = blockIdx.x * 16;
    const int n0   = wave * 16;
    const int lo   = lane & 15;
    const int hi   = lane >> 4;

    int arow = m0 + lo;
    int arow_c = (arow < n) ? arow : (n - 1);       // clamp loads; stores predicated
    const float* ap = X + (long long)arow_c * DIM + 2 * hi;  // A: V0=K+2hi, V1=K+2hi+1
    const float* bp = W + (n0 + lo) + (long long)(2 * hi) * DIM; // B: V0=K(k+2hi), V1=next K row

    float ci = cvec ? cvec[n0 + lo] : 0.0f;          // per-column bias init (all M rows)
    v8f d;
#pragma unroll
    for (int v = 0; v < 8; ++v) d[v] = ci;

#pragma unroll
    for (int k = 0; k < DIM; k += 4) {
        v2f a = *(const v2f*)(ap + k);               // X[m][k+2hi], X[m][k+2hi+1]
        v2f b;
        b.x = bp[(long long)k * DIM];                // W[k+2hi  ][n]
        b.y = bp[(long long)k * DIM + DIM];          // W[k+2hi+1][n]
        d = __builtin_amdgcn_wmma_f32_16x16x4_f32(false, a, false, b,
                                                  (short)0, d, false, false);
    }

    float* yp = Y + (long long)(m0 + 8 * hi) * DIM + n0 + lo;
#pragma unroll
    for (int v = 0; v < 8; ++v) {
        int m = m0 + v + 8 * hi;
        if (m < n) yp[(long long)v * DIM] = d[v];    // C/D: VGPR v -> row v (+8 for hi lanes)
    }
}

// ---------- self-loop init:  acc[i][:] = dinv[i]^2 * xw[i][:]  (float4 per thread) ----------
__global__ void k_selfloop_init(const float* __restrict__ xw, const float* __restrict__ dinv,
                                float* __restrict__ acc, int n) {
    int t = blockIdx.x * blockDim.x + threadIdx.x;   // one per 4 floats
    if (t >= n * 32) return;
    int node = t >> 5;
    float s = dinv[node]; s *= s;
    float4 v = ((const float4*)xw)[t];
    float4 o = make_float4(v.x * s, v.y * s, v.z * s, v.w * s);
    ((float4*)acc)[t] = o;
}

// ---------- edge scatter:  acc[col] += dinv[row]*dinv[col]*xw[row]  (wave per edge) ----------
__global__ void k_edge_scatter(const float* __restrict__ xw,
                               const int* __restrict__ rowi, const int* __restrict__ coli,
                               const float* __restrict__ dinv, float* __restrict__ acc, int e) {
    long long t = (long long)blockIdx.x * blockDim.x + threadIdx.x;
    int eid = (int)(t >> 5);
    if (eid >= e) return;
    int lane = (int)(t & 31);
    int r = rowi[eid], c = coli[eid];
    float norm = dinv[r] * dinv[c];
    float4 v = *(const float4*)(xw + (long long)r * DIM + lane * 4);
    float* dst = acc + (long long)c * DIM + lane * 4;
    atomicAdd(dst + 0, v.x * norm);
    atomicAdd(dst + 1, v.y * norm);
    atomicAdd(dst + 2, v.z * norm);
    atomicAdd(dst + 3, v.w * norm);
}

// ---------- root feature save + constant column vector c for GEMM2 ----------
// root_feat[j] = conv1_out[root][j] = acc[root][j] + b1[j]          (pre-relu, per reference)
// cvec[j]      = sum_k relu(posts[root][k]) * W2[128+k][j]          (broadcast-root term)
__global__ void k_root_and_c(const float* __restrict__ acc, const float* __restrict__ b1,
                             const float* __restrict__ posts, const float* __restrict__ W2,
                             const int* __restrict__ rootPtr,
                             float* __restrict__ root_feat, float* __restrict__ cvec) {
    int j = threadIdx.x;                              // 0..127
    int root = *rootPtr;
    root_feat[j] = acc[(long long)root * DIM + j] + b1[j];
    float s = 0.0f;
    for (int k = 0; k < DIM; ++k)
        s += fmaxf(posts[(long long)root * DIM + k], 0.0f) * W2[(long long)(DIM + k) * DIM + j];
    cvec[j] = s;
}

// ---------- in-place bias + relu:  acc = relu(acc + b1) ----------
__global__ void k_bias_relu(float* __restrict__ acc, const float* __restrict__ b, int n) {
    int t = blockIdx.x * blockDim.x + threadIdx.x;    // one per 4 floats
    if (t >= n * 32) return;
    float4 bb = ((const float4*)b)[t & 31];
    float4 v = ((float4*)acc)[t];
    v.x = fmaxf(v.x + bb.x, 0.0f);
    v.y = fmaxf(v.y + bb.y, 0.0f);
    v.z = fmaxf(v.z + bb.z, 0.0f);
    v.w = fmaxf(v.w + bb.w, 0.0f);
    ((float4*)acc)[t] = v;
}

// ---------- epilogue: feature = [root_feat, relu(acc + b2)]; avg-pool-3 -> out[n x 254] ----------
__global__ __launch_bounds__(256) void k_finalize_pool(
    const float* __restrict__ acc, const float* __restrict__ b2,
    const float* __restrict__ root_feat, float* __restrict__ out, int n) {
    __shared__ float feat[256];
    int i = blockIdx.x;
    int t = threadIdx.x;
    if (t < 128) feat[t] = root_feat[t];
    else         feat[t] = fmaxf(acc[(long long)i * DIM + (t - 128)] + b2[t - 128], 0.0f);
    __syncthreads();
    if (t < 254)
        out[(long long)i * 254 + t] = (feat[t] + feat[t + 1] + feat[t + 2]) * (1.0f / 3.0f);
}

extern "C" void kernel_launch(void* const* d_in, const int* in_sizes, int n_in,
                              void* d_out, int out_size, void* d_ws, size_t ws_size,
                              hipStream_t stream) {
    const float* posts = (const float*)d_in[0];
    const float* W1    = (const float*)d_in[1];
    const float* b1    = (const float*)d_in[2];
    const float* W2    = (const float*)d_in[3];
    const float* b2    = (const float*)d_in[4];
    const int*   edge  = (const int*)d_in[5];
    const int*   rootP = (const int*)d_in[6];

    const int n = in_sizes[0] / DIM;
    const int e = in_sizes[5] / 2;
    const int* rowi = edge;
    const int* coli = edge + e;

    char* ws = (char*)d_ws;
    size_t off = 0;
    float* dinv = (float*)(ws + off); off += (((size_t)n * 4) + 255) & ~(size_t)255;
    float* xw   = (float*)(ws + off); off += (size_t)n * DIM * 4;
    float* acc  = (float*)(ws + off); off += (size_t)n * DIM * 4;
    float* rootf = (float*)(ws + off); off += DIM * 4;
    float* cvec  = (float*)(ws + off);
    float* out = (float*)d_out;

    const int B = 256;
    const int gN   = (n + B - 1) / B;
    const int gE   = (e + B - 1) / B;
    const int gNF4 = (n * 32 + B - 1) / B;                   // n*128/4 threads
    const int gE32 = (int)(((long long)e * 32 + B - 1) / B); // wave per edge
    const int gM   = (n + 15) / 16;

    // degree -> dinv
    k_deg_init <<<gN, B, 0, stream>>>(dinv, n);
    k_deg_count<<<gE, B, 0, stream>>>(coli, dinv, e);
    k_deg_rsqrt<<<gN, B, 0, stream>>>(dinv, n);

    // layer 1
    k_gemm128_wmma <<<gM, B, 0, stream>>>(posts, W1, nullptr, xw, n);
    k_selfloop_init<<<gNF4, B, 0, stream>>>(xw, dinv, acc, n);
    k_edge_scatter <<<gE32, B, 0, stream>>>(xw, rowi, coli, dinv, acc, e);

    // root feature (pre-relu) + constant column vector for layer 2
    k_root_and_c<<<1, DIM, 0, stream>>>(acc, b1, posts, W2, rootP, rootf, cvec);

    // layer 2 (K collapsed 256 -> 128; broadcast-root term folded into cvec)
    k_bias_relu    <<<gNF4, B, 0, stream>>>(acc, b1, n);
    k_gemm128_wmma <<<gM, B, 0, stream>>>(acc, W2, cvec, xw, n);
    k_selfloop_init<<<gNF4, B, 0, stream>>>(xw, dinv, acc, n);
    k_edge_scatter <<<gE32, B, 0, stream>>>(xw, rowi, coli, dinv, acc, e);

    // relu + concat + avg-pool3
    k_finalize_pool<<<n, B, 0, stream>>>(acc, b2, rootf, out, n);
}